// PretrainGNN_82171314307436
// MI455X (gfx1250) — compile-verified
//
#include <hip/hip_runtime.h>
#include <hip/hip_bf16.h>

typedef __attribute__((ext_vector_type(2))) float v2f;
typedef __attribute__((ext_vector_type(8))) float v8f;

#define HID 128
#define NDRUG 5000
#define NPROT 10000
#define NEDGE 250000
#define NLABEL 100000
#define LSM 100
#define CHUNK 250
#define CNNDIM 13600   // 160*85

__device__ __forceinline__ int imin(int a, int b) { return a < b ? a : b; }

// D = A(16x4, f32) x B(4x16, f32) + C(16x16, f32)
// A frag: lane l -> a[r] = A[l&15, 2*(l>>4)+r]
// B frag: lane l -> b[r] = B[2*(l>>4)+r, l&15]
// C/D   : lane l -> c[r] = C[8*(l>>4)+r, l&15]
__device__ __forceinline__ v8f wmma4(v2f a, v2f b, v8f c) {
  return __builtin_amdgcn_wmma_f32_16x16x4_f32(false, a, false, b, (short)0, c,
                                               false, false);
}

// ---------------- permute fc_w: prew[(t*160+c)*128+j] = fc_w[(c*85+t)*128+j]
__global__ void k_permute(const float* __restrict__ fcw, float* __restrict__ prew) {
  int idx = blockIdx.x * blockDim.x + threadIdx.x;
  if (idx >= CNNDIM * HID) return;
  int j = idx & 127;
  int rc = idx >> 7;            // t*160 + c
  int t = rc / 160, c = rc - t * 160;
  prew[idx] = fcw[((c * 85 + t) << 7) + j];
}

// ---------------- conv1: [100,128] tokens->emb, K=512 (k*128+c), M=97, N=40
__global__ void k_conv1(const int* __restrict__ smiles, const float* __restrict__ emb,
                        const float* __restrict__ w, const float* __restrict__ bias,
                        float* __restrict__ x1, int d0) {
  const int tile = blockIdx.y * 8 + (threadIdx.x >> 5);
  if (tile >= 21) return;                       // 7 mtiles x 3 ntiles
  const int mt = tile / 3, nt = tile % 3;
  const int lane = threadIdx.x & 31, l15 = lane & 15, half = lane >> 4;
  const int d = d0 + blockIdx.x;
  const int rowA = imin(mt * 16 + l15, 96);
  const int colB = imin(nt * 16 + l15, 39);
  const int* srow = smiles + d * LSM;
  const float* wrow = w + colB * 512;           // c1_w[o,c,k] -> o*512 + c*4 + k
  v8f acc = {};
  for (int k0 = 0; k0 < 512; k0 += 4) {
    int j = k0 + 2 * half;
    int kseg = j >> 7, c = j & 127;             // pair never crosses 128 boundary
    int tok = srow[rowA + kseg];
    const float* er = emb + tok * HID + c;
    v2f a; a.x = er[0]; a.y = er[1];
    v2f b; b.x = wrow[c * 4 + kseg]; b.y = wrow[(c + 1) * 4 + kseg];
    acc = wmma4(a, b, acc);
  }
  const int col = nt * 16 + l15;
  if (col < 40) {
    float bv = bias[col];
    float* obase = x1 + (size_t)blockIdx.x * 97 * 40;
#pragma unroll
    for (int r = 0; r < 8; ++r) {
      int row = mt * 16 + 8 * half + r;
      if (row < 97) obase[row * 40 + col] = fmaxf(acc[r] + bv, 0.0f);
    }
  }
}

// ---------------- conv2: in [97,40], K=240 (k*40+c), M=92, N=80
__global__ void k_conv2(const float* __restrict__ x1, const float* __restrict__ w,
                        const float* __restrict__ bias, float* __restrict__ x2) {
  const int tile = blockIdx.y * 8 + (threadIdx.x >> 5);
  if (tile >= 30) return;                       // 6 x 5
  const int mt = tile / 5, nt = tile % 5;
  const int lane = threadIdx.x & 31, l15 = lane & 15, half = lane >> 4;
  const int rowA = imin(mt * 16 + l15, 91);
  const int colB = imin(nt * 16 + l15, 79);
  const float* xin = x1 + (size_t)blockIdx.x * 97 * 40;
  const float* wrow = w + colB * 240;           // c2_w[o,c,k] -> o*240 + c*6 + k
  v8f acc = {};
  for (int k0 = 0; k0 < 240; k0 += 4) {
    int j = k0 + 2 * half;
    int kk = j / 40, c = j - kk * 40;           // pair never crosses 40 boundary (j even)
    const float* ir = xin + (rowA + kk) * 40 + c;
    v2f a; a.x = ir[0]; a.y = ir[1];
    v2f b; b.x = wrow[c * 6 + kk]; b.y = wrow[(c + 1) * 6 + kk];
    acc = wmma4(a, b, acc);
  }
  const int col = nt * 16 + l15;
  if (col < 80) {
    float bv = bias[col];
    float* obase = x2 + (size_t)blockIdx.x * 92 * 80;
#pragma unroll
    for (int r = 0; r < 8; ++r) {
      int row = mt * 16 + 8 * half + r;
      if (row < 92) obase[row * 80 + col] = fmaxf(acc[r] + bv, 0.0f);
    }
  }
}

// ---------------- conv3: in [92,80], K=640 (k*80+c), M=85, N=160 -> x3 [d,85*160]
__global__ void k_conv3(const float* __restrict__ x2, const float* __restrict__ w,
                        const float* __restrict__ bias, float* __restrict__ x3) {
  const int tile = blockIdx.y * 8 + (threadIdx.x >> 5);
  if (tile >= 60) return;                       // 6 x 10
  const int mt = tile / 10, nt = tile % 10;
  const int lane = threadIdx.x & 31, l15 = lane & 15, half = lane >> 4;
  const int rowA = imin(mt * 16 + l15, 84);
  const int col = nt * 16 + l15;                // < 160 always
  const float* xin = x2 + (size_t)blockIdx.x * 92 * 80;
  const float* wrow = w + col * 640;            // c3_w[o,c,k] -> o*640 + c*8 + k
  v8f acc = {};
  for (int k0 = 0; k0 < 640; k0 += 4) {
    int j = k0 + 2 * half;
    int kk = j / 80, c = j - kk * 80;
    const float* ir = xin + (rowA + kk) * 80 + c;
    v2f a; a.x = ir[0]; a.y = ir[1];
    v2f b; b.x = wrow[c * 8 + kk]; b.y = wrow[(c + 1) * 8 + kk];
    acc = wmma4(a, b, acc);
  }
  float bv = bias[col];
  float* obase = x3 + (size_t)blockIdx.x * CNNDIM;
#pragma unroll
  for (int r = 0; r < 8; ++r) {
    int row = mt * 16 + 8 * half + r;
    if (row < 85) obase[row * 160 + col] = fmaxf(acc[r] + bv, 0.0f);
  }
}

// ---------------- fc: feat[d0+m, :] = x3[m,:13600] @ prew + fc_b ; M=cnt, N=128
__global__ void k_fc(const float* __restrict__ x3, const float* __restrict__ prew,
                     const float* __restrict__ fcb, float* __restrict__ feat,
                     int d0, int cnt) {
  const int tile = blockIdx.x * 8 + (threadIdx.x >> 5);   // 16 blocks -> 128 tiles
  const int mt = tile >> 3, nt = tile & 7;
  const int lane = threadIdx.x & 31, l15 = lane & 15, half = lane >> 4;
  const int rowA = imin(mt * 16 + l15, cnt - 1);
  const int col = nt * 16 + l15;
  const float* arow = x3 + (size_t)rowA * CNNDIM;
  v8f acc = {};
  for (int k0 = 0; k0 < CNNDIM; k0 += 4) {
    int j = k0 + 2 * half;
    v2f a; a.x = arow[j]; a.y = arow[j + 1];
    v2f b; b.x = prew[(size_t)j * HID + col]; b.y = prew[(size_t)(j + 1) * HID + col];
    acc = wmma4(a, b, acc);
  }
  float bv = fcb[col];
#pragma unroll
  for (int r = 0; r < 8; ++r) {
    int row = mt * 16 + 8 * half + r;
    if (row < cnt) feat[(size_t)(d0 + row) * HID + col] = acc[r] + bv;
  }
}

// ---------------- degree / scatter-add for SAGE mean aggregation
__global__ void k_degree(float* __restrict__ deg, const int* __restrict__ dst) {
  int e = blockIdx.x * blockDim.x + threadIdx.x;
  if (e < NEDGE) atomicAdd(&deg[dst[e]], 1.0f);
}

__global__ void k_scatter(float* __restrict__ agg, const float* __restrict__ x,
                          const int* __restrict__ src, const int* __restrict__ dst) {
  int gid = blockIdx.x * blockDim.x + threadIdx.x;
  int e = gid >> 5;
  if (e >= NEDGE) return;
  int c = (gid & 31) * 4;
  const float* xr = x + (size_t)src[e] * HID + c;
  float* ar = agg + (size_t)dst[e] * HID + c;
  atomicAdd(ar + 0, xr[0]); atomicAdd(ar + 1, xr[1]);
  atomicAdd(ar + 2, xr[2]); atomicAdd(ar + 3, xr[3]);
}

// ---------------- SAGE combine: out = act((agg/deg)@Wl + bl + xdst@Wr [+extra])
__global__ void k_sage(const float* __restrict__ agg, const float* __restrict__ deg,
                       const float* __restrict__ xdst, const float* __restrict__ Wl,
                       const float* __restrict__ bl, const float* __restrict__ Wr,
                       const float* __restrict__ extra, float* __restrict__ out,
                       int M, int do_relu) {
  const int wave = threadIdx.x >> 5;            // 8 waves = 8 N-tiles (N=128)
  const int lane = threadIdx.x & 31, l15 = lane & 15, half = lane >> 4;
  const int rowA = imin(blockIdx.x * 16 + l15, M - 1);
  const int col = wave * 16 + l15;
  const float rdeg = 1.0f / fmaxf(deg[rowA], 1.0f);
  const float* arow = agg + (size_t)rowA * HID;
  const float* xrow = xdst + (size_t)rowA * HID;
  v8f acc = {};
  for (int k0 = 0; k0 < HID; k0 += 4) {
    int j = k0 + 2 * half;
    v2f a; a.x = arow[j] * rdeg; a.y = arow[j + 1] * rdeg;
    v2f b; b.x = Wl[j * HID + col]; b.y = Wl[(j + 1) * HID + col];
    acc = wmma4(a, b, acc);
  }
  for (int k0 = 0; k0 < HID; k0 += 4) {
    int j = k0 + 2 * half;
    v2f a; a.x = xrow[j]; a.y = xrow[j + 1];
    v2f b; b.x = Wr[j * HID + col]; b.y = Wr[(j + 1) * HID + col];
    acc = wmma4(a, b, acc);
  }
  float bv = bl[col];
#pragma unroll
  for (int r = 0; r < 8; ++r) {
    int row = blockIdx.x * 16 + 8 * half + r;
    if (row < M) {
      float v = acc[r] + bv;
      if (extra) v += extra[(size_t)row * HID + col];
      if (do_relu) v = fmaxf(v, 0.0f);
      out[(size_t)row * HID + col] = v;
    }
  }
}

// ---------------- decoder: per 16 labels: H=relu([zd|zp]@W1+b1); score=H@w2+b2
#define HS 132
__global__ void k_dec(const float* __restrict__ zd, const float* __restrict__ zp,
                      const int* __restrict__ esrc, const int* __restrict__ edst,
                      const float* __restrict__ w1, const float* __restrict__ b1,
                      const float* __restrict__ w2, const float* __restrict__ b2,
                      float* __restrict__ scores) {
  __shared__ float Ht[16 * HS];
  const int wave = threadIdx.x >> 5;            // 8 waves = 8 N-tiles of H (N=128)
  const int lane = threadIdx.x & 31, l15 = lane & 15, half = lane >> 4;
  const int e0 = blockIdx.x * 16;
  const int e = e0 + l15;                       // NLABEL % 16 == 0
  const int s = esrc[e], dd = edst[e];
  const int col = wave * 16 + l15;
  const float* zdr = zd + (size_t)s * HID;
  const float* zpr = zp + (size_t)dd * HID;
  v8f acc = {};
  for (int k0 = 0; k0 < 2 * HID; k0 += 4) {
    int j = k0 + 2 * half;
    const float* ar = (j < HID) ? (zdr + j) : (zpr + (j - HID));
    v2f a; a.x = ar[0]; a.y = ar[1];
    v2f b; b.x = w1[j * HID + col]; b.y = w1[(j + 1) * HID + col];
    acc = wmma4(a, b, acc);
  }
  float bv = b1[col];
#pragma unroll
  for (int r = 0; r < 8; ++r) {
    int row = 8 * half + r;
    Ht[row * HS + col] = fmaxf(acc[r] + bv, 0.0f);
  }
  __syncthreads();
  if (threadIdx.x < 16) {
    int row = threadIdx.x;
    float sum = b2[0];
    for (int n = 0; n < HID; ++n) sum += Ht[row * HS + n] * w2[n];
    scores[e0 + row] = sum;
  }
}

extern "C" void kernel_launch(void* const* d_in, const int* in_sizes, int n_in,
                              void* d_out, int out_size, void* d_ws, size_t ws_size,
                              hipStream_t stream) {
  const float* x_drug   = (const float*)d_in[0];
  const float* x_prot   = (const float*)d_in[1];
  const int*   smiles   = (const int*)d_in[2];
  const int*   ei_dp_src= (const int*)d_in[3];
  const int*   ei_dp_dst= (const int*)d_in[4];
  const int*   ei_pd_src= (const int*)d_in[5];
  const int*   ei_pd_dst= (const int*)d_in[6];
  const int*   ell_src  = (const int*)d_in[7];
  const int*   ell_dst  = (const int*)d_in[8];
  const float* emb      = (const float*)d_in[9];
  const float* c1_w = (const float*)d_in[10]; const float* c1_b = (const float*)d_in[11];
  const float* c2_w = (const float*)d_in[12]; const float* c2_b = (const float*)d_in[13];
  const float* c3_w = (const float*)d_in[14]; const float* c3_b = (const float*)d_in[15];
  const float* fc_w = (const float*)d_in[16]; const float* fc_b = (const float*)d_in[17];
  const float* l1_dp_Wl = (const float*)d_in[18]; const float* l1_dp_bl = (const float*)d_in[19];
  const float* l1_dp_Wr = (const float*)d_in[20];
  const float* l1_pd_Wl = (const float*)d_in[21]; const float* l1_pd_bl = (const float*)d_in[22];
  const float* l1_pd_Wr = (const float*)d_in[23];
  const float* l2_dp_Wl = (const float*)d_in[24]; const float* l2_dp_bl = (const float*)d_in[25];
  const float* l2_dp_Wr = (const float*)d_in[26];
  const float* l2_pd_Wl = (const float*)d_in[27]; const float* l2_pd_bl = (const float*)d_in[28];
  const float* l2_pd_Wr = (const float*)d_in[29];
  const float* dec_w1 = (const float*)d_in[30]; const float* dec_b1 = (const float*)d_in[31];
  const float* dec_w2 = (const float*)d_in[32]; const float* dec_b2 = (const float*)d_in[33];

  // workspace layout (floats)
  float* ws       = (float*)d_ws;
  float* prew     = ws;                          // 13600*128 = 1,740,800
  float* x1       = prew + 1740800;              // 250*97*40 =   970,000
  float* x2       = x1 + 970000;                 // 250*92*80 = 1,840,000
  float* x3       = x2 + 1840000;                // 250*13600 = 3,400,000
  float* feat     = x3 + 3400000;                // 5000*128  =   640,000
  float* h_drug   = feat + 640000;               //             640,000
  float* h_prot   = h_drug + 640000;             //           1,280,000
  float* agg_drug = h_prot + 1280000;            //             640,000
  float* agg_prot = agg_drug + 640000;           //           1,280,000
  float* deg_drug = agg_prot + 1280000;          //               5,000
  float* deg_prot = deg_drug + 5000;             //              10,000

  float* z_drug_out = (float*)d_out;             // [5000,128]
  float* z_prot_out = z_drug_out + NDRUG * HID;  // [10000,128]
  float* scores_out = z_drug_out + (NDRUG + NPROT) * HID; // [100000]

  // 0) permute fc weights to [t*160+c] order
  k_permute<<<(CNNDIM * HID + 255) / 256, 256, 0, stream>>>(fc_w, prew);

  // 1) CNN over drugs, chunked (workspace reuse)
  for (int ck = 0; ck < NDRUG / CHUNK; ++ck) {
    int d0 = ck * CHUNK;
    k_conv1<<<dim3(CHUNK, 3), 256, 0, stream>>>(smiles, emb, c1_w, c1_b, x1, d0);
    k_conv2<<<dim3(CHUNK, 4), 256, 0, stream>>>(x1, c2_w, c2_b, x2);
    k_conv3<<<dim3(CHUNK, 8), 256, 0, stream>>>(x2, c3_w, c3_b, x3);
    k_fc<<<16, 256, 0, stream>>>(x3, prew, fc_b, feat, d0, CHUNK);
  }

  // 2) SAGE layer 1
  hipMemsetAsync(agg_prot, 0, (size_t)NPROT * HID * 4, stream);
  hipMemsetAsync(agg_drug, 0, (size_t)NDRUG * HID * 4, stream);
  hipMemsetAsync(deg_prot, 0, (size_t)NPROT * 4, stream);
  hipMemsetAsync(deg_drug, 0, (size_t)NDRUG * 4, stream);
  k_degree<<<(NEDGE + 255) / 256, 256, 0, stream>>>(deg_prot, ei_dp_dst);
  k_degree<<<(NEDGE + 255) / 256, 256, 0, stream>>>(deg_drug, ei_pd_dst);
  k_scatter<<<(NEDGE * 32 + 255) / 256, 256, 0, stream>>>(agg_prot, x_drug, ei_dp_src, ei_dp_dst);
  k_scatter<<<(NEDGE * 32 + 255) / 256, 256, 0, stream>>>(agg_drug, x_prot, ei_pd_src, ei_pd_dst);
  k_sage<<<(NPROT + 15) / 16, 256, 0, stream>>>(agg_prot, deg_prot, x_prot,
      l1_dp_Wl, l1_dp_bl, l1_dp_Wr, nullptr, h_prot, NPROT, 1);
  k_sage<<<(NDRUG + 15) / 16, 256, 0, stream>>>(agg_drug, deg_drug, x_drug,
      l1_pd_Wl, l1_pd_bl, l1_pd_Wr, nullptr, h_drug, NDRUG, 1);

  // 3) SAGE layer 2 (degrees unchanged; re-aggregate hidden states)
  hipMemsetAsync(agg_prot, 0, (size_t)NPROT * HID * 4, stream);
  hipMemsetAsync(agg_drug, 0, (size_t)NDRUG * HID * 4, stream);
  k_scatter<<<(NEDGE * 32 + 255) / 256, 256, 0, stream>>>(agg_prot, h_drug, ei_dp_src, ei_dp_dst);
  k_scatter<<<(NEDGE * 32 + 255) / 256, 256, 0, stream>>>(agg_drug, h_prot, ei_pd_src, ei_pd_dst);
  k_sage<<<(NPROT + 15) / 16, 256, 0, stream>>>(agg_prot, deg_prot, h_prot,
      l2_dp_Wl, l2_dp_bl, l2_dp_Wr, nullptr, z_prot_out, NPROT, 0);
  k_sage<<<(NDRUG + 15) / 16, 256, 0, stream>>>(agg_drug, deg_drug, h_drug,
      l2_pd_Wl, l2_pd_bl, l2_pd_Wr, feat, z_drug_out, NDRUG, 0);   // += CNN feat

  // 4) edge decoder
  k_dec<<<NLABEL / 16, 256, 0, stream>>>(z_drug_out, z_prot_out, ell_src, ell_dst,
                                         dec_w1, dec_b1, dec_w2, dec_b2, scores_out);
}